// GCNResidualBlock_27625229648341
// MI455X (gfx1250) — compile-verified
//
#include <hip/hip_runtime.h>

typedef __attribute__((ext_vector_type(2))) float v2f;
typedef __attribute__((ext_vector_type(8))) float v8f;

#define N_NODES 100000
#define N_EDGES 1600000
#define HID 128

// ---------------- degree / norm kernels ----------------

__global__ __launch_bounds__(256) void k_init_deg(float* __restrict__ deg) {
    int i = blockIdx.x * 256 + threadIdx.x;
    if (i < N_NODES) deg[i] = 1.0f;  // self-loop contributes 1
}

__global__ __launch_bounds__(256) void k_deg(const int* __restrict__ dst,
                                             float* __restrict__ deg) {
    int e = blockIdx.x * 256 + threadIdx.x;
    if (e < N_EDGES) atomicAdd(&deg[dst[e]], 1.0f);
}

__global__ __launch_bounds__(256) void k_dinv(float* __restrict__ deg) {
    int i = blockIdx.x * 256 + threadIdx.x;
    if (i < N_NODES) deg[i] = rsqrtf(deg[i]);  // deg >= 1 always
}

// ---------------- WMMA GEMM: out = f(A @ W) ----------------
// MODE 0: out = A @ W                     (A = x, W = W_gcn)
// MODE 1: A' = relu(A + b_in) on load; out = relu(A'@W + b_out) + xres
template <int MODE>
__global__ __launch_bounds__(256) void k_gemm(
    const float* __restrict__ A, const float* __restrict__ W,
    const float* __restrict__ b_in, const float* __restrict__ b_out,
    const float* __restrict__ xres, float* __restrict__ out) {
    __shared__ float sW[HID * HID];  // 64 KB: full 128x128 weight tile

    const int tid = threadIdx.x;
    // stage W into LDS: 16384 floats / 256 threads = 16 float4 each
    for (int i = tid * 4; i < HID * HID; i += 256 * 4)
        *(float4*)&sW[i] = *(const float4*)&W[i];
    __syncthreads();

    const int wave = tid >> 5;       // 8 waves -> 8 N-tiles of 16
    const int lane = tid & 31;
    const int half = lane >> 4;      // K-pair / row-half select
    const int lm   = lane & 15;
    const int kb   = half * 2;
    const int m0   = blockIdx.x * 16;
    const int n0   = wave * 16;

    const float* Arow = A + (size_t)(m0 + lm) * HID;

    v8f c = {};
    #pragma unroll 4
    for (int kk = 0; kk < HID; kk += 4) {
        v2f a, b;
        if (MODE == 0) {
            a = *(const v2f*)(Arow + kk + kb);  // 8B aligned (kk+kb even)
        } else {
            float a0 = Arow[kk + kb]     + b_in[kk + kb];
            float a1 = Arow[kk + kb + 1] + b_in[kk + kb + 1];
            a.x = a0 > 0.f ? a0 : 0.f;
            a.y = a1 > 0.f ? a1 : 0.f;
        }
        b.x = sW[(kk + kb) * HID + n0 + lm];
        b.y = sW[(kk + kb + 1) * HID + n0 + lm];
        c = __builtin_amdgcn_wmma_f32_16x16x4_f32(false, a, false, b,
                                                  (short)0, c, false, false);
    }

    const int n = n0 + lm;
    #pragma unroll
    for (int v = 0; v < 8; ++v) {
        int m = m0 + v + half * 8;
        float val = c[v];
        if (MODE == 1) {
            val += b_out[n];
            val = val > 0.f ? val : 0.f;
            val += xres[(size_t)m * HID + n];
        }
        out[(size_t)m * HID + n] = val;
    }
}

// ---------------- self-loop init: agg = dinv^2 * xw ----------------

__global__ __launch_bounds__(256) void k_selfloop(const float* __restrict__ dinv,
                                                  const float* __restrict__ xw,
                                                  float* __restrict__ agg) {
    size_t g = (size_t)blockIdx.x * 256 + threadIdx.x;  // one float4 per thread
    size_t node = g >> 5;
    int lane = (int)(g & 31);
    if (node >= N_NODES) return;
    float di = dinv[node];
    float w = di * di;
    float4 v = *(const float4*)(xw + node * HID + lane * 4);
    float4 r = {w * v.x, w * v.y, w * v.z, w * v.w};
    *(float4*)(agg + node * HID + lane * 4) = r;
}

// ---------------- edge scatter: agg[dst] += norm * xw[src] ----------------
// one wave (32 lanes) per edge, float4 per lane -> 128 features

__global__ __launch_bounds__(256) void k_scatter(const int* __restrict__ src,
                                                 const int* __restrict__ dst,
                                                 const float* __restrict__ dinv,
                                                 const float* __restrict__ xw,
                                                 float* __restrict__ agg) {
    size_t g = (size_t)blockIdx.x * 256 + threadIdx.x;
    size_t e = g >> 5;
    int lane = (int)(g & 31);
    if (e >= N_EDGES) return;
    int s = src[e];
    int d = dst[e];
    float norm = dinv[s] * dinv[d];
    float4 v = *(const float4*)(xw + (size_t)s * HID + lane * 4);
    float* p = agg + (size_t)d * HID + lane * 4;
    atomicAdd(p + 0, norm * v.x);
    atomicAdd(p + 1, norm * v.y);
    atomicAdd(p + 2, norm * v.z);
    atomicAdd(p + 3, norm * v.w);
}

// ---------------- launcher ----------------

extern "C" void kernel_launch(void* const* d_in, const int* in_sizes, int n_in,
                              void* d_out, int out_size, void* d_ws, size_t ws_size,
                              hipStream_t stream) {
    const float* x     = (const float*)d_in[0];
    const int*   ei    = (const int*)d_in[1];
    const float* W_gcn = (const float*)d_in[2];
    const float* b_gcn = (const float*)d_in[3];
    const float* W_lin = (const float*)d_in[4];
    const float* b_lin = (const float*)d_in[5];

    const int* src = ei;             // edge_index[0]
    const int* dst = ei + N_EDGES;   // edge_index[1]

    float* out = (float*)d_out;

    // workspace: [0, 400000) = deg/dinv ; [400000, +51.2MB) = agg
    float* dinv = (float*)d_ws;
    float* agg  = (float*)((char*)d_ws + (size_t)N_NODES * sizeof(float));
    float* xw   = out;  // park xw in d_out; fully rewritten by final GEMM

    const int NB_NODE  = (N_NODES + 255) / 256;          // 391
    const int NB_EDGE  = (N_EDGES + 255) / 256;          // 6250
    const int NB_TILE  = N_NODES / 16;                   // 6250 (exact)
    const int NB_SELF  = (N_NODES * 32 + 255) / 256;     // 12500
    const int NB_SCAT  = (int)(((size_t)N_EDGES * 32 + 255) / 256);  // 200000

    k_init_deg<<<NB_NODE, 256, 0, stream>>>(dinv);
    k_deg<<<NB_EDGE, 256, 0, stream>>>(dst, dinv);
    k_dinv<<<NB_NODE, 256, 0, stream>>>(dinv);

    // xw = x @ W_gcn
    k_gemm<0><<<NB_TILE, 256, 0, stream>>>(x, W_gcn, nullptr, nullptr, nullptr, xw);

    // agg = dinv^2 * xw  (self loops), then scatter edges
    k_selfloop<<<NB_SELF, 256, 0, stream>>>(dinv, xw, agg);
    k_scatter<<<NB_SCAT, 256, 0, stream>>>(src, dst, dinv, xw, agg);

    // out = relu( relu(agg + b_gcn) @ W_lin + b_lin ) + x
    k_gemm<1><<<NB_TILE, 256, 0, stream>>>(agg, W_lin, b_gcn, b_lin, x, out);
}